// Spatial_attn_68848325754886
// MI455X (gfx1250) — compile-verified
//
#include <hip/hip_runtime.h>

typedef __attribute__((ext_vector_type(2))) float v2f;
typedef __attribute__((ext_vector_type(8))) float v8f;

#define N_PIX 6400
#define C_CH 8

// ---------------------------------------------------------------------------
// Phase 1: 1x1 conv projections f1,f2,f3 (B,N) + min/max of f1 per batch.
// One block per batch. Deterministic LDS tree reduction (no float atomics).
// ---------------------------------------------------------------------------
__global__ __launch_bounds__(256) void proj_kernel(
    const float* __restrict__ x,
    const float* __restrict__ w1, const float* __restrict__ b1,
    const float* __restrict__ w2, const float* __restrict__ b2,
    const float* __restrict__ w3, const float* __restrict__ b3,
    float* __restrict__ f1g, float* __restrict__ f2g, float* __restrict__ f3g,
    float* __restrict__ f1ext)  // per batch: [max, min]
{
  const int b   = blockIdx.x;
  const int tid = threadIdx.x;
  const float* xb = x + (size_t)b * C_CH * N_PIX;

  float W1[8], W2[8], W3[8];
#pragma unroll
  for (int c = 0; c < 8; ++c) { W1[c] = w1[c]; W2[c] = w2[c]; W3[c] = w3[c]; }
  const float B1 = b1[0], B2 = b2[0], B3 = b3[0];

  float lmax = -3.402823466e38f, lmin = 3.402823466e38f;
  for (int n = tid; n < N_PIX; n += 256) {
    float a1 = B1, a2 = B2, a3 = B3;
#pragma unroll
    for (int c = 0; c < 8; ++c) {
      const float xv = xb[c * N_PIX + n];
      a1 = fmaf(xv, W1[c], a1);
      a2 = fmaf(xv, W2[c], a2);
      a3 = fmaf(xv, W3[c], a3);
    }
    f1g[b * N_PIX + n] = a1;
    f2g[b * N_PIX + n] = a2;
    f3g[b * N_PIX + n] = a3;
    lmax = fmaxf(lmax, a1);
    lmin = fminf(lmin, a1);
  }

  __shared__ float smax[256];
  __shared__ float smin[256];
  smax[tid] = lmax; smin[tid] = lmin;
  __syncthreads();
  for (int s = 128; s > 0; s >>= 1) {
    if (tid < s) {
      smax[tid] = fmaxf(smax[tid], smax[tid + s]);
      smin[tid] = fminf(smin[tid], smin[tid + s]);
    }
    __syncthreads();
  }
  if (tid == 0) { f1ext[b * 2 + 0] = smax[0]; f1ext[b * 2 + 1] = smin[0]; }
}

// ---------------------------------------------------------------------------
// Phase 2: per m: den = sum_n exp(s*f1[n]-rm), num = sum_n f3[n]*exp(...),
//          gate = sigmoid(num/den), out = x * gate.
// One wave handles 16 m-rows; lanes L and L+16 share m = m0+(L&15) and split
// the n range. Final lane-pair reduction of (pden,pnum) is done by a single
// V_WMMA_F32_16X16X4_F32 with a 0/1 selector B-matrix:
//   D[m][0] = pden_h0[m] + pden_h1[m],  D[m][1] = pnum_h0[m] + pnum_h1[m].
// ---------------------------------------------------------------------------
__global__ __launch_bounds__(64) void attn_kernel(
    const float* __restrict__ x,
    const float* __restrict__ f1g, const float* __restrict__ f2g,
    const float* __restrict__ f3g, const float* __restrict__ f1ext,
    float* __restrict__ out)
{
  const int blocksPerBatch = N_PIX / 32;  // 200 (32 m per block: 2 waves x 16)
  const int b       = blockIdx.x / blocksPerBatch;
  const int m_block = (blockIdx.x % blocksPerBatch) * 32;
  const int wave  = threadIdx.x >> 5;
  const int lane  = threadIdx.x & 31;
  const int laneN = lane & 15;
  const int half  = lane >> 4;
  const int m0    = m_block + wave * 16;
  const int m     = m0 + laneN;

  const float s     = f2g[b * N_PIX + m];
  const float f1max = f1ext[b * 2 + 0];
  const float f1min = f1ext[b * 2 + 1];
  const float nrm   = -((s >= 0.0f) ? s * f1max : s * f1min);  // -rowmax[m]

  const v2f* __restrict__ f1v = (const v2f*)(f1g + b * N_PIX);
  const v2f* __restrict__ f3v = (const v2f*)(f3g + b * N_PIX);

  float pden = 0.0f, pnum = 0.0f;
  // lanes 0-15 handle n%4 in {0,1}; lanes 16-31 handle n%4 in {2,3}
  for (int n = 0; n < N_PIX; n += 4) {
    const int idx = (n >> 1) + half;       // float2 index -> elements {k0,k0+1}
    const v2f f1p = f1v[idx];
    const v2f f3p = f3v[idx];
    const float e0 = __expf(fmaf(s, f1p.x, nrm));
    const float e1 = __expf(fmaf(s, f1p.y, nrm));
    pden += e0;
    pden += e1;
    pnum = fmaf(f3p.x, e0, pnum);
    pnum = fmaf(f3p.y, e1, pnum);
  }

  // A (16x4 f32): lane L<16 -> A[L][0]=a.x, A[L][1]=a.y ; lane L+16 -> A[L][2]=a.x, A[L][3]=a.y
  v2f a;  a.x = pden;                    a.y = pnum;
  // B (4x16 f32): B[k][0]=1 for even/odd selector -> b.x=(laneN==0), b.y=(laneN==1)
  v2f bsel; bsel.x = (laneN == 0) ? 1.0f : 0.0f;
  bsel.y = (laneN == 1) ? 1.0f : 0.0f;
  v8f cacc = {};
  // 8 args: (neg_a, A, neg_b, B, c_mod, C, reuse_a, reuse_b) -> v_wmma_f32_16x16x4_f32
  cacc = __builtin_amdgcn_wmma_f32_16x16x4_f32(
      false, a, false, bsel, (short)0, cacc, false, false);

  // C/D layout: VGPR r, lanes 0-15 = D[r][lane]; lanes 16-31 = D[r+8][lane-16].
  // den[m] lives in column 0 (lanes 0 & 16), num[m] in column 1 (lanes 1 & 17).
  __shared__ float sden[2][16];
  __shared__ float snum[2][16];
  __shared__ float sgate[2][16];
  if (laneN <= 1) {
    float* dst = (laneN == 0) ? sden[wave] : snum[wave];
#pragma unroll
    for (int r = 0; r < 8; ++r) dst[half * 8 + r] = cacc[r];
  }
  __syncthreads();
  if (lane < 16) {
    const float t = snum[wave][laneN] / sden[wave][laneN];
    sgate[wave][laneN] = 1.0f / (1.0f + __expf(-t));
  }
  __syncthreads();

  const float* xb = x   + (size_t)b * C_CH * N_PIX;
  float*       ob = out + (size_t)b * C_CH * N_PIX;
#pragma unroll
  for (int j = 0; j < 4; ++j) {
    const int i  = lane + j * 32;   // 0..127 covers 8 channels x 16 m
    const int c  = i >> 4;
    const int mi = i & 15;
    const int p  = c * N_PIX + m0 + mi;
    ob[p] = xb[p] * sgate[wave][mi];
  }
}

// ---------------------------------------------------------------------------
extern "C" void kernel_launch(void* const* d_in, const int* in_sizes, int n_in,
                              void* d_out, int out_size, void* d_ws, size_t ws_size,
                              hipStream_t stream) {
  (void)in_sizes; (void)n_in; (void)out_size; (void)ws_size;
  const float* x  = (const float*)d_in[0];
  const float* w1 = (const float*)d_in[1];
  const float* b1 = (const float*)d_in[2];
  const float* w2 = (const float*)d_in[3];
  const float* b2 = (const float*)d_in[4];
  const float* w3 = (const float*)d_in[5];
  const float* b3 = (const float*)d_in[6];
  float* out = (float*)d_out;

  float* ws    = (float*)d_ws;
  float* f1g   = ws;                 // 2*6400
  float* f2g   = ws + 2 * N_PIX;     // 2*6400
  float* f3g   = ws + 4 * N_PIX;     // 2*6400
  float* f1ext = ws + 6 * N_PIX;     // 4 floats

  proj_kernel<<<2, 256, 0, stream>>>(x, w1, b1, w2, b2, w3, b3,
                                     f1g, f2g, f3g, f1ext);
  attn_kernel<<<(2 * N_PIX) / 32, 64, 0, stream>>>(x, f1g, f2g, f3g, f1ext, out);
}